// ProximalFiedlerRefinement_6356551598163
// MI455X (gfx1250) — compile-verified
//
#include <hip/hip_runtime.h>
#include <hip/hip_bf16.h>

#define TPB 256
#define TAU 0.5f
#define EPS 1e-12f

typedef __attribute__((ext_vector_type(2))) float v2f;
typedef __attribute__((ext_vector_type(8))) float v8f;

// ---------------------------------------------------------------------------
// Zero the accumulator vector and the norm scalar for this iteration.
// ---------------------------------------------------------------------------
__global__ __launch_bounds__(TPB) void pf_zero_kernel(float* __restrict__ v,
                                                      float* __restrict__ norm_acc,
                                                      int n) {
    int i = blockIdx.x * blockDim.x + threadIdx.x;
    if (i < n) v[i] = 0.0f;
    if (i == 0) *norm_acc = 0.0f;
}

// ---------------------------------------------------------------------------
// COO SpMV: vout[r] += vals[e] * vin[c].  Edge lists streamed as B128 loads
// (4 edges per lane), grid-stride persistent loop with global_prefetch of the
// next tile so HBM streaming overlaps the L2-resident random gather + atomic
// scatter.  vin/vout are 2MB each -> resident in the 192MB L2.
// ---------------------------------------------------------------------------
__global__ __launch_bounds__(TPB) void pf_spmv_kernel(
    const float4* __restrict__ vals4, const int4* __restrict__ rows4,
    const int4* __restrict__ cols4, const float* __restrict__ vin,
    float* __restrict__ vout, int nnz4) {
    int stride = gridDim.x * blockDim.x;
    for (int i = blockIdx.x * blockDim.x + threadIdx.x; i < nnz4; i += stride) {
        int ip = i + stride;
        if (ip < nnz4) {
            // gfx1250: lowers to global_prefetch_b8 (speculative, pulls full lines)
            __builtin_prefetch(&vals4[ip], 0, 3);
            __builtin_prefetch(&rows4[ip], 0, 3);
            __builtin_prefetch(&cols4[ip], 0, 3);
        }
        float4 v = vals4[i];
        int4   c = cols4[i];
        int4   r = rows4[i];
        float x0 = vin[c.x];
        float x1 = vin[c.y];
        float x2 = vin[c.z];
        float x3 = vin[c.w];
        // Non-returning float atomics -> GLOBAL_ATOMIC_ADD_F32 (STOREcnt path)
        atomicAdd(&vout[r.x], v.x * x0);
        atomicAdd(&vout[r.y], v.y * x1);
        atomicAdd(&vout[r.z], v.z * x2);
        atomicAdd(&vout[r.w], v.w * x3);
    }
}

// Scalar tail for NNZ not divisible by 4 (robustness; NNZ=16M is divisible).
__global__ __launch_bounds__(64) void pf_spmv_tail_kernel(
    const float* __restrict__ vals, const int* __restrict__ rows,
    const int* __restrict__ cols, const float* __restrict__ vin,
    float* __restrict__ vout, int start, int nnz) {
    int i = start + blockIdx.x * blockDim.x + threadIdx.x;
    if (i < nnz) atomicAdd(&vout[rows[i]], vals[i] * vin[cols[i]]);
}

// ---------------------------------------------------------------------------
// Wave32 sum across all 32 lanes using V_WMMA_F32_16X16X4_F32.
// A: VGPR0 = per-lane partial p, VGPR1 = 0   (covers all 32 partials exactly
//    once among the 16x4 A elements; remaining elements are 0).
// B: all ones (layout-independent since every element is 1).
// => D(m,n) = rowsum_m for all n, so any column of D sums to sum(p).
// Column n of D lives in VGPRs 0..7 of lane n (M=0..7) and lane n+16 (M=8..15),
// so per-lane sum of the 8 D regs + one shfl_xor(16) add yields the wave total
// in every lane.  EXEC must be all-ones: callers reach this unconditionally.
// ---------------------------------------------------------------------------
__device__ inline float pf_wave_sum_wmma(float p) {
    v2f a;
    a.x = p;
    a.y = 0.0f;
    v2f b;
    b.x = 1.0f;
    b.y = 1.0f;
    v8f c = {};
    v8f d = __builtin_amdgcn_wmma_f32_16x16x4_f32(
        /*neg_a=*/false, a, /*neg_b=*/false, b,
        /*c_mod=*/(short)0, c, /*reuse_a=*/false, /*reuse_b=*/false);
    float s = d[0] + d[1] + d[2] + d[3] + d[4] + d[5] + d[6] + d[7];
    s += __shfl_xor(s, 16);
    return s;
}

// ---------------------------------------------------------------------------
// v <- v - tau*sign(v)  (in place), and accumulate sum(v*v) into *norm_acc.
// ---------------------------------------------------------------------------
__global__ __launch_bounds__(TPB) void pf_shrink_reduce_kernel(
    float* __restrict__ v, float* __restrict__ norm_acc, int n) {
    __shared__ float wave_sums[TPB / 32];
    int i = blockIdx.x * blockDim.x + threadIdx.x;
    float x = 0.0f;
    if (i < n) {
        x = v[i];
        float s = (x > 0.0f) ? 1.0f : ((x < 0.0f) ? -1.0f : 0.0f);
        x = x - TAU * s;
        v[i] = x;
    }
    float p = x * x;
    float wsum = pf_wave_sum_wmma(p);  // reached by all lanes uniformly

    int lane = threadIdx.x & 31;
    int wave = threadIdx.x >> 5;
    if (lane == 0) wave_sums[wave] = wsum;
    __syncthreads();
    if (threadIdx.x == 0) {
        float bsum = 0.0f;
        for (int w = 0; w < TPB / 32; ++w) bsum += wave_sums[w];
        atomicAdd(norm_acc, bsum);
    }
}

// ---------------------------------------------------------------------------
// dst <- src / max(sqrt(norm), eps)
// ---------------------------------------------------------------------------
__global__ __launch_bounds__(TPB) void pf_normalize_kernel(
    const float* __restrict__ src, float* __restrict__ dst,
    const float* __restrict__ norm_acc, int n) {
    __shared__ float inv;
    if (threadIdx.x == 0) {
        float nrm = sqrtf(*norm_acc);
        inv = 1.0f / fmaxf(nrm, EPS);
    }
    __syncthreads();
    int i = blockIdx.x * blockDim.x + threadIdx.x;
    if (i < n) dst[i] = src[i] * inv;
}

// ---------------------------------------------------------------------------
extern "C" void kernel_launch(void* const* d_in, const int* in_sizes, int n_in,
                              void* d_out, int out_size, void* d_ws, size_t ws_size,
                              hipStream_t stream) {
    const float* scores = (const float*)d_in[0];
    const float* vals   = (const float*)d_in[1];
    const int*   rows   = (const int*)d_in[2];
    const int*   cols   = (const int*)d_in[3];
    float*       out    = (float*)d_out;

    const int n   = in_sizes[0];  // 500000
    const int nnz = in_sizes[1];  // 16000000

    // Workspace layout: two ping-pong N-vectors + norm scalar (~4MB).
    float* buf0     = (float*)d_ws;
    float* buf1     = buf0 + n;
    float* norm_acc = buf1 + n;

    const int nblk_n   = (n + TPB - 1) / TPB;
    const int nnz4     = nnz / 4;
    const int tail     = nnz - nnz4 * 4;
    int spmv_blocks    = 3072;  // persistent grid-stride; plenty of wave32 occupancy
    if ((long long)spmv_blocks * TPB > (long long)nnz4)
        spmv_blocks = (nnz4 + TPB - 1) / TPB;
    if (spmv_blocks < 1) spmv_blocks = 1;

    const float4* vals4 = (const float4*)vals;
    const int4*   rows4 = (const int4*)rows;
    const int4*   cols4 = (const int4*)cols;

    const int ITERS = 10;
    const float* vin = scores;
    float* bufs[2] = {buf0, buf1};

    for (int t = 0; t < ITERS; ++t) {
        float* vout = bufs[t & 1];

        pf_zero_kernel<<<nblk_n, TPB, 0, stream>>>(vout, norm_acc, n);

        if (nnz4 > 0)
            pf_spmv_kernel<<<spmv_blocks, TPB, 0, stream>>>(vals4, rows4, cols4,
                                                            vin, vout, nnz4);
        if (tail > 0)
            pf_spmv_tail_kernel<<<1, 64, 0, stream>>>(vals, rows, cols, vin, vout,
                                                      nnz4 * 4, nnz);

        pf_shrink_reduce_kernel<<<nblk_n, TPB, 0, stream>>>(vout, norm_acc, n);

        float* dst = (t == ITERS - 1) ? out : vout;
        pf_normalize_kernel<<<nblk_n, TPB, 0, stream>>>(vout, dst, norm_acc, n);

        vin = dst;
    }
}